// MultiScaleSTGATBlock_10831907520909
// MI455X (gfx1250) — compile-verified
//
#include <hip/hip_runtime.h>
#include <hip/hip_bf16.h>

typedef __attribute__((ext_vector_type(2))) float v2f;
typedef __attribute__((ext_vector_type(8))) float v8f;

#define B_  32
#define C_  64
#define T_  256
#define V_  25
#define N_  (B_ * T_)          // 8192 (b,t) slices

__device__ inline v2f lds2(const float* p) { return *(const v2f*)p; }

// ---------------------------------------------------------------------------
// Kernel 1 (fused GAT): per (b,t) slice
//   Wh = h @ W  (WMMA, LDS-staged, kept in LDS)
//   e1/e2 dots, masked leaky-relu softmax
//   g = elu(att @ Wh)  (WMMA from LDS), store G in (B, V, T, C) layout.
// Block = 256 threads (8 waves). All fragments are aligned ds_load_b64 with
// even padded strides (68 / 36) -> conflict-free, no exec-branch guards.
// ---------------------------------------------------------------------------
#define XS 68   // xsT/wT row stride (even, bank-spread)
#define WS 36   // whT/att row stride (even, bank-spread)

__global__ __launch_bounds__(256) void k_gat_fused(const float* __restrict__ x,
                                                   const float* __restrict__ W,
                                                   const float* __restrict__ a_gat,
                                                   const float* __restrict__ adj,
                                                   float* __restrict__ G) {
    __shared__ float xsT[32 * XS];   // [v][c]  (v padded 25->32 with zeros)
    __shared__ float wT [64 * XS];   // [d][c]
    __shared__ float whT[64 * WS];   // [d][v]
    __shared__ float att[32 * WS];   // [u][v]  (pad cols zero)
    __shared__ float e1s[32];
    __shared__ float e2s[32];

    int n = blockIdx.x;
    int b = n >> 8;
    int t = n & 255;
    int tid = threadIdx.x;

    // ---- stage x slice, transposed: xsT[v][c] = x[b, c, t, v] ----
    const float* xb = x + (size_t)b * (C_ * T_ * V_) + (size_t)t * V_;
    for (int idx = tid; idx < 32 * 64; idx += 256) {
        int v = idx & 31, c = idx >> 5;
        xsT[v * XS + c] = (v < V_) ? xb[(size_t)c * (T_ * V_) + v] : 0.0f;
    }
    // ---- stage W transposed: wT[d][c] = W[c, d] ----
    for (int idx = tid; idx < 64 * 64; idx += 256) {
        int d = idx & 63, c = idx >> 6;
        wT[d * XS + c] = W[c * 64 + d];
    }
    __syncthreads();

    int wave  = tid >> 5;
    int l     = tid & 31;
    int l15   = l & 15;
    int lhalf = l >> 4;

    // ---- Wh GEMM: M=32(v) x N=64(d), K=64(c); wave -> (m0, n0) tile ----
    {
        int m0 = (wave & 1) * 16;
        int n0 = (wave >> 1) * 16;
        v8f acc = {};
        #pragma unroll
        for (int kk = 0; kk < 16; ++kk) {
            int c0 = kk * 4 + lhalf * 2;
            v2f A  = lds2(&xsT[(m0 + l15) * XS + c0]);
            v2f Bv = lds2(&wT [(n0 + l15) * XS + c0]);
            acc = __builtin_amdgcn_wmma_f32_16x16x4_f32(false, A, false, Bv,
                                                        (short)0, acc, false, false);
        }
        #pragma unroll
        for (int r = 0; r < 8; ++r)
            whT[(n0 + l15) * WS + (m0 + r + lhalf * 8)] = acc[r];
    }
    __syncthreads();

    // ---- e1[u] = Wh[u,:]@a1 ; e2[u] = Wh[u,:]@a2 (wave 0) ----
    if (tid < 32) {
        float e1 = 0.f, e2 = 0.f;
        if (tid < V_) {
            #pragma unroll 8
            for (int d = 0; d < 64; ++d) {
                float w = whT[d * WS + tid];
                e1 += w * a_gat[d];
                e2 += w * a_gat[64 + d];
            }
        }
        e1s[tid] = e1;
        e2s[tid] = e2;
    }
    __syncthreads();

    // ---- masked leaky-relu + softmax rows (wave 0) ----
    if (tid < 32) {
        if (tid < V_) {
            float row[V_];
            float m = -3.4e38f;
            #pragma unroll
            for (int vv = 0; vv < V_; ++vv) {
                float e = e1s[tid] + e2s[vv];
                e = (e >= 0.f) ? e : 0.2f * e;               // leaky_relu 0.2
                e = (adj[tid * V_ + vv] > 0.f) ? e : -9e15f; // mask
                row[vv] = e;
                m = fmaxf(m, e);
            }
            float s = 0.f;
            #pragma unroll
            for (int vv = 0; vv < V_; ++vv) {
                float p = expf(row[vv] - m);
                row[vv] = p;
                s += p;
            }
            float inv = 1.f / s;
            #pragma unroll
            for (int vv = 0; vv < V_; ++vv) att[tid * WS + vv] = row[vv] * inv;
            #pragma unroll
            for (int vv = V_; vv < 32; ++vv) att[tid * WS + vv] = 0.f;
        } else {
            for (int vv = 0; vv < 32; ++vv) att[tid * WS + vv] = 0.f;
        }
    }
    __syncthreads();

    // ---- g = elu(att @ Wh): M=32(u) x N=64(d), K=32(v, padded) ----
    {
        int mt = wave & 1;
        int nt = wave >> 1;
        v8f acc = {};
        #pragma unroll
        for (int kk = 0; kk < 8; ++kk) {
            int k0 = kk * 4 + lhalf * 2;
            v2f A  = lds2(&att[(mt * 16 + l15) * WS + k0]);
            v2f Bv = lds2(&whT[(nt * 16 + l15) * WS + k0]);
            acc = __builtin_amdgcn_wmma_f32_16x16x4_f32(false, A, false, Bv,
                                                        (short)0, acc, false, false);
        }
        #pragma unroll
        for (int r = 0; r < 8; ++r) {
            int u = mt * 16 + r + lhalf * 8;
            if (u < V_) {
                float g = acc[r];
                g = (g > 0.f) ? g : (expf(g) - 1.f);        // elu
                int d = nt * 16 + l15;
                G[(((size_t)b * V_ + u) * T_ + t) * 64 + d] = g;
            }
        }
    }
}

// ---------------------------------------------------------------------------
// Kernel 2: fused  BN1 -> ReLU -> conv{3,5,7} -> BN2 -> concat -> 1x1 conv.
// Block = (t-tile of 32, v, b), 8 waves. Weight A-frags are aligned float2
// global loads; BN2 & conv bias folded into z; 1x1 fusion GEMM reads LDS.
// ---------------------------------------------------------------------------
#define TT 32
#define HALO 3
#define GROW 41   // gtile row stride (odd -> conflict-free b32 columns)
#define ZROW 33   // z row stride

template <int Ki>
__device__ inline void branch_gemm(int i, int m0, int n0, int l15, int lhalf,
                                   const float* __restrict__ w,
                                   const float* gt, const float* s1, const float* h1,
                                   const float* s2, const float* o2, float* zt) {
    const int pad  = (Ki - 1) / 2;
    const int Ktot = 64 * Ki;
    v8f acc = {};
    #pragma unroll 4
    for (int kk = 0; kk < Ktot / 4; ++kk) {
        int ka = kk * 4 + lhalf * 2;
        v2f A = *(const v2f*)(w + (size_t)(m0 + l15) * Ktot + ka); // 8B aligned
        int ci0 = ka / Ki,       k0 = ka - ci0 * Ki;
        int ci1 = (ka + 1) / Ki, k1 = (ka + 1) - ci1 * Ki;
        float g0 = gt[ci0 * GROW + (n0 + l15) + k0 - pad + HALO];
        float g1 = gt[ci1 * GROW + (n0 + l15) + k1 - pad + HALO];
        v2f Bv;
        Bv.x = fmaxf(0.f, g0 * s1[i * 64 + ci0] + h1[i * 64 + ci0]);
        Bv.y = fmaxf(0.f, g1 * s1[i * 64 + ci1] + h1[i * 64 + ci1]);
        acc = __builtin_amdgcn_wmma_f32_16x16x4_f32(false, A, false, Bv,
                                                    (short)0, acc, false, false);
    }
    #pragma unroll
    for (int r = 0; r < 8; ++r) {
        int cb = m0 + r + lhalf * 8;
        zt[(i * 64 + cb) * ZROW + (n0 + l15)] =
            acc[r] * s2[i * 64 + cb] + o2[i * 64 + cb];
    }
}

__global__ __launch_bounds__(256) void k_conv(
    const float* __restrict__ G,
    const float* __restrict__ w3, const float* __restrict__ b3,
    const float* __restrict__ w5, const float* __restrict__ b5,
    const float* __restrict__ w7, const float* __restrict__ b7,
    const float* __restrict__ bn1g, const float* __restrict__ bn1b,
    const float* __restrict__ bn1m, const float* __restrict__ bn1v,
    const float* __restrict__ bn2g, const float* __restrict__ bn2b,
    const float* __restrict__ bn2m, const float* __restrict__ bn2v,
    const float* __restrict__ wf, const float* __restrict__ bf,
    float* __restrict__ out) {

    __shared__ float gt[64 * GROW];
    __shared__ float zt[192 * ZROW];
    __shared__ float s1[192], h1[192], s2[192], o2[192];

    int t0 = blockIdx.x * TT;
    int v  = blockIdx.y;
    int b  = blockIdx.z;
    int tid = threadIdx.x;

    if (tid < 192) {
        int i = tid >> 6, c = tid & 63;
        float sv = bn1g[tid] * rsqrtf(bn1v[tid] + 1e-5f);
        s1[tid] = sv;
        h1[tid] = bn1b[tid] - bn1m[tid] * sv;
        float s2v = bn2g[tid] * rsqrtf(bn2v[tid] + 1e-5f);
        s2[tid] = s2v;
        const float* bi = (i == 0) ? b3 : (i == 1) ? b5 : b7;
        o2[tid] = s2v * bi[c] + (bn2b[tid] - bn2m[tid] * s2v);
    }

    const float* gb = G + (((size_t)b * V_ + v) * T_) * 64;
    for (int idx = tid; idx < (TT + 2 * HALO) * 64; idx += 256) {
        int ci = idx & 63, xx = idx >> 6;
        int tsrc = t0 + xx - HALO;
        gt[ci * GROW + xx] =
            (tsrc >= 0 && tsrc < T_) ? gb[(size_t)tsrc * 64 + ci] : 0.f;
    }
    __syncthreads();

    int wave  = tid >> 5;
    int l     = tid & 31;
    int l15   = l & 15;
    int lhalf = l >> 4;
    int m0 = (wave >> 1) * 16;  // 4 M-tiles (channels)
    int n0 = (wave & 1) * 16;   // 2 N-tiles (time)

    branch_gemm<3>(0, m0, n0, l15, lhalf, w3, gt, s1, h1, s2, o2, zt);
    branch_gemm<5>(1, m0, n0, l15, lhalf, w5, gt, s1, h1, s2, o2, zt);
    branch_gemm<7>(2, m0, n0, l15, lhalf, w7, gt, s1, h1, s2, o2, zt);
    __syncthreads();

    // final 1x1: out[co,t] = bf[co] + sum_j wf[co,j] * z[j][t];  K = 192
    v8f acc = {};
    #pragma unroll 4
    for (int kk = 0; kk < 48; ++kk) {
        int ka = kk * 4 + lhalf * 2;
        v2f A = *(const v2f*)(wf + (size_t)(m0 + l15) * 192 + ka);
        v2f Bv;
        Bv.x = zt[ka * ZROW + n0 + l15];
        Bv.y = zt[(ka + 1) * ZROW + n0 + l15];
        acc = __builtin_amdgcn_wmma_f32_16x16x4_f32(false, A, false, Bv,
                                                    (short)0, acc, false, false);
    }
    #pragma unroll
    for (int r = 0; r < 8; ++r) {
        int co = m0 + r + lhalf * 8;
        int t  = t0 + n0 + l15;
        out[(((size_t)b * 64 + co) * T_ + t) * V_ + v] = acc[r] + bf[co];
    }
}

// ---------------------------------------------------------------------------
extern "C" void kernel_launch(void* const* d_in, const int* in_sizes, int n_in,
                              void* d_out, int out_size, void* d_ws, size_t ws_size,
                              hipStream_t stream) {
    (void)in_sizes; (void)n_in; (void)out_size; (void)ws_size;

    const float* x    = (const float*)d_in[0];
    const float* adj  = (const float*)d_in[1];
    const float* Wgat = (const float*)d_in[2];
    const float* agat = (const float*)d_in[3];
    const float* bn1g = (const float*)d_in[4];
    const float* bn1b = (const float*)d_in[5];
    const float* bn1m = (const float*)d_in[6];
    const float* bn1v = (const float*)d_in[7];
    const float* w3   = (const float*)d_in[8];
    const float* b3   = (const float*)d_in[9];
    const float* w5   = (const float*)d_in[10];
    const float* b5   = (const float*)d_in[11];
    const float* w7   = (const float*)d_in[12];
    const float* b7   = (const float*)d_in[13];
    const float* bn2g = (const float*)d_in[14];
    const float* bn2b = (const float*)d_in[15];
    const float* bn2m = (const float*)d_in[16];
    const float* bn2v = (const float*)d_in[17];
    const float* wf   = (const float*)d_in[18];
    const float* bf   = (const float*)d_in[19];
    float* outp = (float*)d_out;

    float* G = (float*)d_ws;   // B*V*T*64 floats (~52.4 MB)

    k_gat_fused<<<N_, 256, 0, stream>>>(x, Wgat, agat, adj, G);
    dim3 grid(T_ / TT, V_, B_);
    k_conv<<<grid, 256, 0, stream>>>(G, w3, b3, w5, b5, w7, b7,
                                     bn1g, bn1b, bn1m, bn1v,
                                     bn2g, bn2b, bn2m, bn2v,
                                     wf, bf, outp);
}